// Attn_82480551952797
// MI455X (gfx1250) — compile-verified
//
#include <hip/hip_runtime.h>
#include <math.h>

// ---------------------------------------------------------------------------
// Problem constants (from reference)
// ---------------------------------------------------------------------------
#define BB     4
#define SS     512
#define HH     768
#define NHEAD  4
#define DHEAD  192
#define MAXW   8
#define NSPAN  4096            // N = S * MAXW
#define BN     (BB * NSPAN)    // 16384 span rows
#define WDIM   64
#define INNER  3072
#define INDIM  1600            // H + WD + H
#define KVLD   1536            // K(768) | V(768) packed per token

#define LDSPAD 36              // floats per LDS tile row (16B-aligned, conflict-free)

static const float QK_SCALE = 0.07216878364870323f;  // 1/sqrt(192)

typedef __attribute__((ext_vector_type(2))) float v2f;
typedef __attribute__((ext_vector_type(8))) float v8f;
typedef __attribute__((ext_vector_type(4))) int   v4i;

// fp32 WMMA: D[16,16] = A[16,4] x B[4,16] + C   (matrix pipe, full fp32)
static __device__ __forceinline__ v8f wmma_f32(v2f a, v2f b, v8f c) {
  return __builtin_amdgcn_wmma_f32_16x16x4_f32(
      /*neg_a=*/false, a, /*neg_b=*/false, b,
      /*c_mod=*/(short)0, c, /*reuse_a=*/false, /*reuse_b=*/false);
}

// CDNA5 async memory->LDS path (ASYNCcnt-tracked, no VGPR staging).
#if defined(__has_builtin)
#if __has_builtin(__builtin_amdgcn_global_load_async_to_lds_b128) && \
    __has_builtin(__builtin_amdgcn_s_wait_asynccnt)
#define USE_ASYNC_LDS 1
#endif
#endif

#ifdef USE_ASYNC_LDS
// builtin signature (from hipcc diagnostic): param0 = int4 in AS1 ("__device__"),
// param1 = int4 in AS3, then imm offset + imm cpol.
static __device__ __forceinline__ void async_cp16(const float* gsrc, float* ldst) {
  __builtin_amdgcn_global_load_async_to_lds_b128(
      (__attribute__((address_space(1))) v4i*)(gsrc),
      (__attribute__((address_space(3))) v4i*)(ldst), 0, 0);
}
#endif

// ---------------------------------------------------------------------------
// Kernel 1: qkv_cls = in_proj(cls_embedding)  -> [3*768] (q | k_cls | v_cls)
// ---------------------------------------------------------------------------
__global__ void qkv_cls_kernel(const float* __restrict__ w,
                               const float* __restrict__ bias,
                               const float* __restrict__ cls_emb,
                               float* __restrict__ qkv) {
  int wid  = (blockIdx.x * blockDim.x + threadIdx.x) >> 5;
  int lane = threadIdx.x & 31;
  if (wid >= 3 * HH) return;
  const float* wr = w + (size_t)wid * HH;
  float acc = 0.f;
  for (int j = lane; j < HH; j += 32) acc += wr[j] * cls_emb[j];
  #pragma unroll
  for (int o = 16; o; o >>= 1) acc += __shfl_xor(acc, o, 32);
  if (lane == 0) qkv[wid] = acc + bias[wid];
}

// ---------------------------------------------------------------------------
// Kernel 2: cls_score[h] = scale * <q_h, kcls_h>   (4 heads, one block)
// ---------------------------------------------------------------------------
__global__ void cls_score_kernel(const float* __restrict__ qkv,
                                 float* __restrict__ cls_score) {
  int h = threadIdx.x >> 5, lane = threadIdx.x & 31;
  const float* q = qkv;
  const float* k = qkv + HH;
  float acc = 0.f;
  for (int d = lane; d < DHEAD; d += 32) acc += q[h * DHEAD + d] * k[h * DHEAD + d];
  #pragma unroll
  for (int o = 16; o; o >>= 1) acc += __shfl_xor(acc, o, 32);
  if (lane == 0) cls_score[h] = acc * QK_SCALE;
}

// ---------------------------------------------------------------------------
// Kernel 3 (shared GEMM): C[M,N] = act(A[M,K] @ W[N,K]^T + bias)
// block = 256 thr (8 waves), block tile 64x128, wave tile 32x32 (2x2 wmma).
// K staged 32 at a time into DOUBLE-BUFFERED LDS; the copy is
// GLOBAL_LOAD_ASYNC_TO_LDS_B128 overlapped with the WMMA compute of the
// other buffer (one barrier per K-tile).
// Rows padded to 36 floats: 16B-aligned b128 targets + conflict-free v2f reads.
// All problem dims are exact tile multiples (no edge handling, EXEC all-ones).
// ---------------------------------------------------------------------------
__global__ __launch_bounds__(256) void gemm_wmma_kernel(
    const float* __restrict__ A, int lda,
    const float* __restrict__ W, int K,
    const float* __restrict__ bias,
    float* __restrict__ C, int ldc, int relu) {
  __shared__ float As[2][64 * LDSPAD];
  __shared__ float Ws[2][128 * LDSPAD];

  const int tid  = threadIdx.x;
  const int lane = tid & 31;
  const int wave = tid >> 5;
  const int wm   = wave >> 2;          // 0..1 : M direction
  const int wn   = wave & 3;           // 0..3 : N direction
  const int half = lane >> 4;          // hi/lo 16 lanes
  const int l16  = lane & 15;
  const size_t rowBase = (size_t)blockIdx.y * 64;
  const size_t colBase = (size_t)blockIdx.x * 128;

  // per-thread staging coordinates: A tile 64x32 = 2 b128/thread,
  // W tile 128x32 = 4 b128/thread (coalesced: 8 threads per row)
  const int sRow = tid >> 3;           // 0..31
  const int sC4  = (tid & 7) << 2;     // 0,4,...,28
  const float* aSrc0 = A + (rowBase + sRow) * (size_t)lda + sC4;
  const float* aSrc1 = A + (rowBase + sRow + 32) * (size_t)lda + sC4;
  const float* wSrc0 = W + (colBase + sRow) * (size_t)K + sC4;
  const float* wSrc1 = W + (colBase + sRow + 32) * (size_t)K + sC4;
  const float* wSrc2 = W + (colBase + sRow + 64) * (size_t)K + sC4;
  const float* wSrc3 = W + (colBase + sRow + 96) * (size_t)K + sC4;
  const int aDst = sRow * LDSPAD + sC4;
  const int wDst = sRow * LDSPAD + sC4;

  auto issue_tile = [&](int kc, int buf) {
#ifdef USE_ASYNC_LDS
    async_cp16(aSrc0 + kc, &As[buf][aDst]);
    async_cp16(aSrc1 + kc, &As[buf][aDst + 32 * LDSPAD]);
    async_cp16(wSrc0 + kc, &Ws[buf][wDst]);
    async_cp16(wSrc1 + kc, &Ws[buf][wDst + 32 * LDSPAD]);
    async_cp16(wSrc2 + kc, &Ws[buf][wDst + 64 * LDSPAD]);
    async_cp16(wSrc3 + kc, &Ws[buf][wDst + 96 * LDSPAD]);
#else
    *(float4*)&As[buf][aDst]               = *(const float4*)(aSrc0 + kc);
    *(float4*)&As[buf][aDst + 32 * LDSPAD] = *(const float4*)(aSrc1 + kc);
    *(float4*)&Ws[buf][wDst]               = *(const float4*)(wSrc0 + kc);
    *(float4*)&Ws[buf][wDst + 32 * LDSPAD] = *(const float4*)(wSrc1 + kc);
    *(float4*)&Ws[buf][wDst + 64 * LDSPAD] = *(const float4*)(wSrc2 + kc);
    *(float4*)&Ws[buf][wDst + 96 * LDSPAD] = *(const float4*)(wSrc3 + kc);
#endif
  };

  v8f acc00 = {}, acc01 = {}, acc10 = {}, acc11 = {};

  const int ntiles = K >> 5;
  issue_tile(0, 0);
  for (int t = 0; t < ntiles; ++t) {
    const int buf = t & 1;
#ifdef USE_ASYNC_LDS
    __builtin_amdgcn_s_wait_asynccnt(0);   // this wave's copies landed in LDS
#endif
    __syncthreads();                        // whole tile visible; prior readers done
    if (t + 1 < ntiles) issue_tile((t + 1) << 5, buf ^ 1);  // overlap with compute

    const float* as = As[buf];
    const float* ws = Ws[buf];
    #pragma unroll
    for (int kk = 0; kk < 8; ++kk) {
      int kb = kk * 4 + half * 2;  // A frag: lanes 0-15 K{0,1}, 16-31 K{2,3}
      v2f a0 = *(const v2f*)&as[(wm * 32 +      l16) * LDSPAD + kb];
      v2f a1 = *(const v2f*)&as[(wm * 32 + 16 + l16) * LDSPAD + kb];
      v2f b0 = *(const v2f*)&ws[(wn * 32 +      l16) * LDSPAD + kb];
      v2f b1 = *(const v2f*)&ws[(wn * 32 + 16 + l16) * LDSPAD + kb];
      acc00 = wmma_f32(a0, b0, acc00);
      acc01 = wmma_f32(a0, b1, acc01);
      acc10 = wmma_f32(a1, b0, acc10);
      acc11 = wmma_f32(a1, b1, acc11);
    }
  }

  // epilogue: D layout = VGPR i -> row (i + 8*half), col = lane%16
  int col0 = (int)colBase + wn * 32 + l16;
  int col1 = col0 + 16;
  float bv0 = bias ? bias[col0] : 0.f;
  float bv1 = bias ? bias[col1] : 0.f;
  #pragma unroll
  for (int mi = 0; mi < 2; ++mi) {
    v8f va = mi ? acc10 : acc00;
    v8f vb = mi ? acc11 : acc01;
    size_t rbase = rowBase + wm * 32 + mi * 16 + half * 8;
    #pragma unroll
    for (int i = 0; i < 8; ++i) {
      float x0 = va[i] + bv0;
      float x1 = vb[i] + bv1;
      if (relu) { x0 = fmaxf(x0, 0.f); x1 = fmaxf(x1, 0.f); }
      C[(rbase + i) * (size_t)ldc + col0] = x0;
      C[(rbase + i) * (size_t)ldc + col1] = x1;
    }
  }
}

// ---------------------------------------------------------------------------
// Kernel 4: token scores[b,h,s] = scale * <q_h, K[b,s,h]>   (wave per (b,s))
// ---------------------------------------------------------------------------
__global__ void token_score_kernel(const float* __restrict__ qkv,
                                   const float* __restrict__ kv,
                                   float* __restrict__ scores) {
  int wid  = (blockIdx.x * blockDim.x + threadIdx.x) >> 5;
  int lane = threadIdx.x & 31;
  if (wid >= BB * SS) return;
  const float* krow = kv + (size_t)wid * KVLD;   // K half: cols 0..767
  int b = wid / SS, s = wid % SS;
  for (int h = 0; h < NHEAD; ++h) {
    float acc = 0.f;
    for (int d = lane; d < DHEAD; d += 32) acc += qkv[h * DHEAD + d] * krow[h * DHEAD + d];
    #pragma unroll
    for (int o = 16; o; o >>= 1) acc += __shfl_xor(acc, o, 32);
    if (lane == 0) scores[((size_t)b * NHEAD + h) * SS + s] = acc * QK_SCALE;
  }
}

// ---------------------------------------------------------------------------
// Kernel 5: per-span 9-entry softmax + weighted V sum -> ctx[row,768];
// also writes width-embedding and cls_reps slices of feats[row,1600].
// ---------------------------------------------------------------------------
__global__ __launch_bounds__(256) void span_ctx_kernel(
    const int* __restrict__ span_ids, const unsigned char* __restrict__ span_masks,
    const int* __restrict__ span_widths, const float* __restrict__ scores,
    const float* __restrict__ cls_score, const float* __restrict__ qkv,
    const float* __restrict__ kv, const float* __restrict__ width_table,
    const float* __restrict__ cls_reps, float* __restrict__ ctx,
    float* __restrict__ feats) {
  int row = blockIdx.x;              // 0..BN-1
  int b   = row / NSPAN;
  int t   = threadIdx.x;
  __shared__ float wgt[NHEAD][MAXW + 1];   // normalized; [h][0]=CLS, [h][1+j]=token j

  int start = span_ids[2 * row];
  int weff  = span_ids[2 * row + 1] - start;     // span width, 1..8
  if (!span_masks[row]) weff = 0;                // masked span -> only CLS valid

  if (t < NHEAD) {
    float sc[MAXW + 1];
    sc[0] = cls_score[t];
    const float* srow = scores + ((size_t)b * NHEAD + t) * SS;
    for (int j = 0; j < weff; ++j) sc[1 + j] = srow[start + j];
    float m = sc[0];
    for (int j = 1; j <= weff; ++j) m = fmaxf(m, sc[j]);
    float e[MAXW + 1];
    float sum = 0.f;
    for (int j = 0; j <= weff; ++j) { e[j] = expf(sc[j] - m); sum += e[j]; }
    float inv = 1.f / sum;
    for (int j = 0; j <= MAXW; ++j) wgt[t][j] = (j <= weff) ? e[j] * inv : 0.f;
  }
  __syncthreads();

  const float* vcls  = qkv + 2 * HH;
  const float* vbase = kv + ((size_t)(b * SS + start)) * KVLD + HH;  // V half
  float* crow = ctx + (size_t)row * HH;
  #pragma unroll
  for (int rep = 0; rep < 3; ++rep) {
    int dim = t + rep * 256;
    int h   = dim / DHEAD;
    float acc = wgt[h][0] * vcls[dim];
    #pragma unroll
    for (int j = 0; j < MAXW; ++j)           // zero weights past weff -> branchless
      acc += wgt[h][1 + j] * vbase[(size_t)j * KVLD + dim];
    crow[dim] = acc;
  }

  float* frow = feats + (size_t)row * INDIM;
  if (t < WDIM) frow[HH + t] = width_table[span_widths[row] * WDIM + t];
  #pragma unroll
  for (int rep = 0; rep < 3; ++rep) {
    int dim = t + rep * 256;
    frow[HH + WDIM + dim] = cls_reps[(size_t)b * HH + dim];
  }
}

// ---------------------------------------------------------------------------
// Launcher
// ---------------------------------------------------------------------------
extern "C" void kernel_launch(void* const* d_in, const int* in_sizes, int n_in,
                              void* d_out, int out_size, void* d_ws, size_t ws_size,
                              hipStream_t stream) {
  (void)in_sizes; (void)n_in; (void)out_size; (void)ws_size;
  const float*         token_reps    = (const float*)d_in[0];
  const int*           span_ids      = (const int*)d_in[1];
  const unsigned char* span_masks    = (const unsigned char*)d_in[2];  // numpy bool
  const float*         cls_reps      = (const float*)d_in[3];
  const int*           span_widths   = (const int*)d_in[4];
  const float*         cls_embedding = (const float*)d_in[5];
  const float*         in_proj_w     = (const float*)d_in[6];
  const float*         in_proj_b     = (const float*)d_in[7];
  const float*         out_proj_w    = (const float*)d_in[8];
  const float*         out_proj_b    = (const float*)d_in[9];
  const float*         width_table   = (const float*)d_in[10];
  const float*         w1            = (const float*)d_in[11];
  const float*         b1            = (const float*)d_in[12];
  const float*         w2            = (const float*)d_in[13];
  const float*         b2            = (const float*)d_in[14];
  float* out = (float*)d_out;

  // workspace layout (floats, 64-float aligned)
  float* ws        = (float*)d_ws;
  float* qkv       = ws;                                   // 2304
  float* cls_score = ws + 2304;                            // 4 (padded)
  float* scores    = ws + 2368;                            // B*NH*S = 8192
  float* kvbuf     = ws + 10560;                           // 2048*1536
  float* feats     = kvbuf + (size_t)(BB * SS) * KVLD;     // BN*1600
  float* h1        = feats + (size_t)BN * INDIM;           // BN*3072
  float* ctx       = h1;   // alias: ctx is dead before h1 is produced

  // 1) cls projections
  qkv_cls_kernel<<<dim3((3 * HH + 7) / 8), 256, 0, stream>>>(
      in_proj_w, in_proj_b, cls_embedding, qkv);

  // 2) CLS key score per head
  cls_score_kernel<<<1, 128, 0, stream>>>(qkv, cls_score);

  // 3) token K|V = token_reps @ [wk;wv]^T + [bk;bv]   [2048,1536]
  gemm_wmma_kernel<<<dim3(KVLD / 128, (BB * SS) / 64), 256, 0, stream>>>(
      token_reps, HH, in_proj_w + (size_t)HH * HH, HH, in_proj_b + HH,
      kvbuf, KVLD, /*relu=*/0);

  // 4) per-token attention scores
  token_score_kernel<<<dim3((BB * SS) / 8), 256, 0, stream>>>(qkv, kvbuf, scores);

  // 5) span softmax + context; width-emb + cls_reps slices of feats
  span_ctx_kernel<<<dim3(BN), 256, 0, stream>>>(
      span_ids, span_masks, span_widths, scores, cls_score, qkv, kvbuf,
      width_table, cls_reps, ctx, feats);

  // 6) out_proj -> feats[:, 0:768]  (ldc = 1600)
  gemm_wmma_kernel<<<dim3(HH / 128, BN / 64), 256, 0, stream>>>(
      ctx, HH, out_proj_w, HH, out_proj_b, feats, INDIM, /*relu=*/0);

  // 7) FFN1 + ReLU: h1 = relu(feats @ w1^T + b1)   [16384,3072]
  gemm_wmma_kernel<<<dim3(INNER / 128, BN / 64), 256, 0, stream>>>(
      feats, INDIM, w1, INDIM, b1, h1, INNER, /*relu=*/1);

  // 8) FFN2: out = h1 @ w2^T + b2   [16384,768]
  gemm_wmma_kernel<<<dim3(HH / 128, BN / 64), 256, 0, stream>>>(
      h1, INNER, w2, INNER, b2, out, HH, /*relu=*/0);
}